// PrimitiveAttentiveAtlasEncoder_27195732918752
// MI455X (gfx1250) — compile-verified
//
#include <hip/hip_runtime.h>

// Problem constants (match reference)
constexpr int BATCH = 16384;
constexpr int LD    = 16;    // latent dim D
constexpr int KD    = 64;    // key dim K
constexpr int NCH   = 64;    // num charts
constexpr int RNK   = 8;     // low-rank
constexpr int BPB   = 16;    // batch elements per block (== WMMA M)
constexpr int THREADS = 512; // 16 waves (wave32)
constexpr int MPITCH  = 66;  // row pitch for M: 8B-aligned rows, conflict-free columns
constexpr int NITER   = 24;  // fixed-point solve iterations (rho~0.23 -> <1e-15)
constexpr float ONE_EPS = 1.001f; // 1 + TRANSPORT_EPS

typedef __attribute__((ext_vector_type(2))) float v2f;
typedef __attribute__((ext_vector_type(8))) float v8f;

__global__ __launch_bounds__(THREADS)
void atlas_router_kernel(const float* __restrict__ z,
                         const float* __restrict__ feat,
                         const float* __restrict__ Wz,
                         const float* __restrict__ bz,
                         const float* __restrict__ Wf,
                         const float* __restrict__ bf,
                         const float* __restrict__ Gu,
                         const float* __restrict__ Gv,
                         const float* __restrict__ Gout,
                         const float* __restrict__ Wt,
                         const float* __restrict__ cq,
                         float* __restrict__ out)
{
  // ~297 KB static LDS: fits CDNA5's 320 KB/WGP workgroup allocation
  __shared__ float sM[BPB * KD * MPITCH];  // 270,336 B : per-batch M = 0.25*(s - s^T)
  __shared__ float sZ[BPB * LD];           // z tile (WMMA A operand source)
  __shared__ float sF[BPB * KD];
  __shared__ float sQ[BPB * KD];
  __shared__ float sW[2 * BPB * KD];       // ping-pong solve vector
  __shared__ float sV[BPB * KD];
  __shared__ float sSc[BPB * NCH];
  __shared__ float sU[BPB * RNK];
  __shared__ float sG[BPB * RNK];
  __shared__ float sInvTau[BPB];
  __shared__ float sMax[BPB];
  __shared__ float sISum[BPB];

  const int tid  = threadIdx.x;
  const int wid  = tid >> 5;
  const int lane = tid & 31;
  const int half = lane >> 4;   // WMMA lane-half
  const int l16  = lane & 15;
  const int blockBase = blockIdx.x * BPB;

  // Warm L2 for Wt (256 KB, reused by all 1024 blocks) -> global_prefetch_b8
  __builtin_prefetch(&Wt[tid * 128], 0, 1);

  // ---- stage z and features tiles into LDS ----
  if (tid < BPB * LD) {
    int b = tid >> 4, d = tid & 15;
    sZ[b * LD + d] = z[(blockBase + b) * LD + d];
  }
  for (int idx = tid; idx < BPB * KD; idx += THREADS) {
    int b = idx >> 6, k = idx & 63;
    sF[b * KD + k] = feat[(blockBase + b) * KD + k];
  }
  __syncthreads();

  // ---- low-rank factors u_r = Gu[r].z, g_r = Gv[r].z  and metric temperature ----
  if (tid < BPB * RNK) {
    int b = tid >> 3, r = tid & 7;
    float u = 0.f, g = 0.f;
    for (int d = 0; d < LD; ++d) {
      float zz = sZ[b * LD + d];
      u = fmaf(Gu[r * LD + d], zz, u);
      g = fmaf(Gv[r * LD + d], zz, g);
    }
    sU[b * RNK + r] = u;
    sG[b * RNK + r] = g;
  }
  if (tid < BPB) {
    float r2 = 0.f;
    for (int d = 0; d < LD; ++d) { float zz = sZ[tid * LD + d]; r2 = fmaf(zz, zz, r2); }
    float denom = fmaxf(1.0f - r2, 0.001f);          // TAU_DENOM_MIN
    float tau = fmaxf(8.0f * denom * 0.5f, 0.01f);   // sqrt(64)=8, TAU_MIN
    sInvTau[tid] = 1.0f / tau;
  }
  __syncthreads();

  // ---- preload WMMA A fragments (z rows) and tile-0 B fragments ----
  // A 16x4 f32 fragment: VGPR0 = K={0|2} per lane-half, VGPR1 = K={1|3}; rows M=l16.
  // B 4x16 fragment mirrors A; C/D: VGPR g holds rows {g, g+8} across lane halves.
  const float* zrow = &sZ[l16 * LD + 2 * half];
  v2f afr[4];
  #pragma unroll
  for (int kk = 0; kk < 4; ++kk) afr[kk] = *(const v2f*)(zrow + 4 * kk);

  const float* wbase = &Wt[(wid * 256 + l16) * LD + 2 * half];
  v2f curB[4];
  #pragma unroll
  for (int kk = 0; kk < 4; ++kk) curB[kk] = *(const v2f*)(wbase + 4 * kk);

  // ---- q = z@Wz^T + bz + f@Wf^T + bf + ((zGu^T)*(zGv^T))@Gout (overlaps Wt loads) ----
  for (int idx = tid; idx < BPB * KD; idx += THREADS) {
    int b = idx >> 6, k = idx & 63;
    float acc = bz[k] + bf[k];
    const float4* wzr = (const float4*)&Wz[k * LD];
    const float4* zr  = (const float4*)&sZ[b * LD];
    #pragma unroll
    for (int d4 = 0; d4 < LD / 4; ++d4) {
      float4 a = wzr[d4], bv = zr[d4];
      acc = fmaf(a.x, bv.x, acc); acc = fmaf(a.y, bv.y, acc);
      acc = fmaf(a.z, bv.z, acc); acc = fmaf(a.w, bv.w, acc);
    }
    const float4* wfr = (const float4*)&Wf[k * KD];
    const float4* fr  = (const float4*)&sF[b * KD];
    #pragma unroll
    for (int j4 = 0; j4 < KD / 4; ++j4) {
      float4 a = wfr[j4], bv = fr[j4];
      acc = fmaf(a.x, bv.x, acc); acc = fmaf(a.y, bv.y, acc);
      acc = fmaf(a.z, bv.z, acc); acc = fmaf(a.w, bv.w, acc);
    }
    #pragma unroll
    for (int r = 0; r < RNK; ++r)
      acc = fmaf(sU[b * RNK + r] * sG[b * RNK + r], Gout[r * KD + k], acc);
    sQ[idx] = acc;
  }

  // ---- S(16x4096) = Ztile(16x16) @ Wt^T(16x4096) via v_wmma_f32_16x16x4_f32 ----
  // 2-stage software pipeline: tile t+1 B-fragments load while tile t WMMAs execute.
  for (int t = 0; t < 16; ++t) {
    const int tn = (t + 1) & 15;
    const float* wnext = &Wt[(wid * 256 + tn * 16 + l16) * LD + 2 * half];
    v2f nxtB[4];
    #pragma unroll
    for (int kk = 0; kk < 4; ++kk) nxtB[kk] = *(const v2f*)(wnext + 4 * kk);

    v8f c = {};
    #pragma unroll
    for (int kk = 0; kk < 4; ++kk)                    // chain K=16 as 4 x (K=4)
      c = __builtin_amdgcn_wmma_f32_16x16x4_f32(
              false, afr[kk], false, curB[kk], (short)0, c, false, false);

    const int p0 = wid * 256 + t * 16;                // output column base
    const int i  = p0 >> 6;                           // row of 64x64 skew source
    const int j  = (p0 & 63) + l16;                   // col
    const int mb = 8 * half;
    #pragma unroll
    for (int g = 0; g < 8; ++g)
      sM[(mb + g) * (KD * MPITCH) + i * MPITCH + j] = c[g];

    #pragma unroll
    for (int kk = 0; kk < 4; ++kk) curB[kk] = nxtB[kk];
  }
  __syncthreads();

  // ---- in-place M = 0.25*(s - s^T)  (== 0.5*skew); pair (i,j)/(j,i) owned by task (b,i) ----
  for (int task = tid; task < BPB * KD; task += THREADS) {
    int b = task >> 6, i = task & 63;
    float* Mb = &sM[b * (KD * MPITCH)];
    Mb[i * MPITCH + i] = 0.f;
    for (int j = i + 1; j < KD; ++j) {
      float a  = Mb[i * MPITCH + j];
      float bt = Mb[j * MPITCH + i];
      float m  = 0.25f * (a - bt);
      Mb[i * MPITCH + j] =  m;
      Mb[j * MPITCH + i] = -m;
    }
  }
  __syncthreads();

  // ---- solve ((1+eps)I - M) w = q by fixed point: w <- (q + M w)/(1+eps) ----
  // (Uses A^T = B for antisymmetric skew: scores need only v = U^T q = A (B^{-1} q))
  const float inv1e = 1.0f / ONE_EPS;
  for (int idx = tid; idx < BPB * KD; idx += THREADS) sW[idx] = sQ[idx] * inv1e;
  __syncthreads();
  int cur = 0;
  for (int it = 0; it < NITER; ++it) {
    const int nxt = cur ^ 1;
    for (int task = tid; task < BPB * KD; task += THREADS) {
      int b = task >> 6, i = task & 63;
      const v2f* Mr = (const v2f*)&sM[b * (KD * MPITCH) + i * MPITCH]; // 8B-aligned (MPITCH even)
      const v2f* wv = (const v2f*)&sW[cur * BPB * KD + b * KD];
      float acc = sQ[task];
      #pragma unroll
      for (int j2 = 0; j2 < KD / 2; ++j2) {
        v2f m = Mr[j2], w = wv[j2];
        acc = fmaf(m.x, w.x, acc);
        acc = fmaf(m.y, w.y, acc);
      }
      sW[nxt * BPB * KD + task] = acc * inv1e;
    }
    cur = nxt;
    __syncthreads();
  }

  // ---- v = (1+eps) w + M w ----
  for (int task = tid; task < BPB * KD; task += THREADS) {
    int b = task >> 6, i = task & 63;
    const v2f* Mr = (const v2f*)&sM[b * (KD * MPITCH) + i * MPITCH];
    const v2f* wv = (const v2f*)&sW[cur * BPB * KD + b * KD];
    float acc = ONE_EPS * sW[cur * BPB * KD + b * KD + i];
    #pragma unroll
    for (int j2 = 0; j2 < KD / 2; ++j2) {
      v2f m = Mr[j2], w = wv[j2];
      acc = fmaf(m.x, w.x, acc);
      acc = fmaf(m.y, w.y, acc);
    }
    sV[task] = acc;
  }
  __syncthreads();

  // ---- scores[b,n] = cq[n] . v[b]  (cq is L2-hot: 16 KB shared by all blocks) ----
  for (int task = tid; task < BPB * NCH; task += THREADS) {
    int b = task >> 6, n = task & 63;
    const float4* cr = (const float4*)&cq[n * KD];
    const float4* vv = (const float4*)&sV[b * KD];
    float acc = 0.f;
    #pragma unroll
    for (int k4 = 0; k4 < KD / 4; ++k4) {
      float4 a = cr[k4], bv = vv[k4];
      acc = fmaf(a.x, bv.x, acc); acc = fmaf(a.y, bv.y, acc);
      acc = fmaf(a.z, bv.z, acc); acc = fmaf(a.w, bv.w, acc);
    }
    sSc[task] = acc;
  }
  __syncthreads();

  // ---- per-row softmax stats + argmax (first-max, matching jnp.argmax) ----
  if (tid < BPB) {
    const float it = sInvTau[tid];
    float mx = -3.4e38f; int am = 0;
    for (int n = 0; n < NCH; ++n) {
      float x = sSc[tid * NCH + n] * it;
      if (x > mx) { mx = x; am = n; }
    }
    float s = 0.f;
    for (int n = 0; n < NCH; ++n) s += __expf(sSc[tid * NCH + n] * it - mx);
    sMax[tid]  = mx;
    sISum[tid] = 1.0f / s;
    out[(size_t)BATCH * NCH + (blockBase + tid)] = (float)am;  // K_chart
  }
  __syncthreads();

  for (int task = tid; task < BPB * NCH; task += THREADS) {
    int b = task >> 6, n = task & 63;
    out[(size_t)(blockBase + b) * NCH + n] =
        __expf(sSc[task] * sInvTau[b] - sMax[b]) * sISum[b];
  }
}

extern "C" void kernel_launch(void* const* d_in, const int* in_sizes, int n_in,
                              void* d_out, int out_size, void* d_ws, size_t ws_size,
                              hipStream_t stream) {
  (void)in_sizes; (void)n_in; (void)out_size; (void)d_ws; (void)ws_size;
  const float* z    = (const float*)d_in[0];
  const float* feat = (const float*)d_in[1];
  const float* Wz   = (const float*)d_in[2];
  const float* bz   = (const float*)d_in[3];
  const float* Wf   = (const float*)d_in[4];
  const float* bf   = (const float*)d_in[5];
  const float* Gu   = (const float*)d_in[6];
  const float* Gv   = (const float*)d_in[7];
  const float* Gout = (const float*)d_in[8];
  const float* Wt   = (const float*)d_in[9];
  const float* cq   = (const float*)d_in[10];
  float* out = (float*)d_out;

  dim3 grid(BATCH / BPB);   // 1024 workgroups
  dim3 block(THREADS);      // 512 threads = 16 wave32 waves
  atlas_router_kernel<<<grid, block, 0, stream>>>(
      z, feat, Wz, bz, Wf, bf, Gu, Gv, Gout, Wt, cq, out);
}